// RotaryPositionalEmbedding_64707977281874
// MI455X (gfx1250) — compile-verified
//
#include <hip/hip_runtime.h>

// RoPE, bandwidth-optimal formulation for MI455X (gfx1250).
// x: [4,16,4096,128] f32; token_positions: [4096] i64; table: [4096,128,128] f32.
// Only the 2x2-block diagonal of the table is used: per position, 64 cos + 64 sin.
// Traffic = 134MB in + 134MB out (+~2MB unique table) -> ~11.5us at 23.3 TB/s.

typedef float v4f __attribute__((ext_vector_type(4)));

#define SEQ_LEN 4096
#define DK      128
#define NVEC    64        // 4 batches * 16 heads

__global__ __launch_bounds__(256) void rope_gfx1250_kernel(
    const float* __restrict__ x,
    const long long* __restrict__ token_positions,
    const float* __restrict__ table,
    float* __restrict__ out)
{
    // Dynamic LDS (starts at LDS offset 0): smem[0:64)=cos_j, smem[64:128)=sin_j
    extern __shared__ float smem[];

    const int tid = threadIdx.x;
    const int s   = blockIdx.x;

    // Per-block uniform: position index and table base for this position.
    const long long p = token_positions[s];
    const float* tb = table + (size_t)p * (size_t)(DK * DK);   // +p*16384 floats

    // ---- Stage cos/sin via CDNA5 async gather to LDS (ASYNCcnt path) ----
    // c_j = T[p, 2j, 2j]     -> float offset j*258
    // s_j = T[p, 2j+1, 2j]   -> float offset j*258 + 128
    // Threads 0..63 gather cos, 64..127 gather sin. 4 full waves issue one
    // global_load_async_to_lds_b32 each (32 lanes * 4 waves = 128 values).
    if (tid < 128) {
        const unsigned j    = (unsigned)(tid & 63);
        const unsigned voff = (j * 258u + ((tid & 64) ? 128u : 0u)) * 4u; // byte off
        const unsigned ldsb = (unsigned)tid * 4u;                         // LDS byte addr
        asm volatile("global_load_async_to_lds_b32 %0, %1, %2"
                     :
                     : "v"(ldsb), "v"(voff), "s"(tb)
                     : "memory");
    }
    asm volatile("s_wait_asynccnt 0" ::: "memory");
    __syncthreads();

    // ---- Streaming rotate: 2048 float4 per block, 8 per thread ----
    const size_t srow = (size_t)s * DK;
    #pragma unroll
    for (int k = 0; k < 8; ++k) {
        const int t  = tid + k * 256;        // 0..2047
        const int v  = t >> 5;               // which (b,h) vector: 0..63
        const int d4 = t & 31;               // float4 index within the 128-dim row
        const size_t f = (size_t)v * (size_t)(SEQ_LEN * DK) + srow + (size_t)d4 * 4;

        // Non-temporal: x/out are touched exactly once; keep L2 for the table.
        v4f xv = __builtin_nontemporal_load((const v4f*)(x + f));

        const int j0 = d4 * 2;
        const int j1 = j0 + 1;
        const float c0 = smem[j0];
        const float s0 = smem[64 + j0];
        const float c1 = smem[j1];
        const float s1 = smem[64 + j1];

        v4f r;
        r.x = c0 * xv.x - s0 * xv.y;
        r.y = s0 * xv.x + c0 * xv.y;
        r.z = c1 * xv.z - s1 * xv.w;
        r.w = s1 * xv.z + c1 * xv.w;

        __builtin_nontemporal_store(r, (v4f*)(out + f));
    }
}

extern "C" void kernel_launch(void* const* d_in, const int* in_sizes, int n_in,
                              void* d_out, int out_size, void* d_ws, size_t ws_size,
                              hipStream_t stream) {
    (void)in_sizes; (void)n_in; (void)out_size; (void)d_ws; (void)ws_size;

    const float*     x     = (const float*)d_in[0];
    const long long* tpos  = (const long long*)d_in[1];   // int64 token positions
    const float*     table = (const float*)d_in[2];
    float*           out   = (float*)d_out;

    // One block per sequence position; 128 floats of dynamic LDS for cos/sin.
    rope_gfx1250_kernel<<<SEQ_LEN, 256, 128 * sizeof(float), stream>>>(
        x, tpos, table, out);
}